// ObjectAffinityFeatures_55697135894756
// MI455X (gfx1250) — compile-verified
//
#include <hip/hip_runtime.h>
#include <hip/hip_bf16.h>

// Problem constants (from reference setup_inputs)
#define BB 8
#define CC 3
#define HH 128
#define WW_ 128
#define DD 132
#define PP (HH*WW_)      // 16384 spatial positions
#define P1 (PP+1)        // +1 appended zero row (index target H*W)
#define NN 2048
#define KK 32

typedef __attribute__((ext_vector_type(2))) float v2f;
typedef __attribute__((ext_vector_type(8))) float v8f;

// ---------------------------------------------------------------------------
// Kernel 1: 3x3 SAME conv + L2 normalize over W (per (b,d,h) row) + store
// transposed as feat[b, h*W+w, d]  (row-major rows of 132 floats)
// One block of 128 threads = one (b,d,h) row, thread = w.
// ---------------------------------------------------------------------------
__global__ __launch_bounds__(128)
void conv_norm_kernel(const float* __restrict__ img,
                      const float* __restrict__ conv_w,
                      const float* __restrict__ conv_b,
                      float* __restrict__ feat) {
  int blk = blockIdx.x;
  int h = blk % HH;
  int d = (blk / HH) % DD;
  int b = blk / (HH * DD);
  int w = threadIdx.x;

  // 27 weights for this output channel (uniform across block)
  float wgt[27];
  #pragma unroll
  for (int i = 0; i < 27; ++i) wgt[i] = conv_w[d * 27 + i];

  float acc = conv_b[d];
  #pragma unroll
  for (int c = 0; c < CC; ++c) {
    #pragma unroll
    for (int kh = 0; kh < 3; ++kh) {
      int hh = h + kh - 1;
      if (hh < 0 || hh >= HH) continue;
      const float* row = img + (((size_t)b * CC + c) * HH + hh) * WW_;
      #pragma unroll
      for (int kw = 0; kw < 3; ++kw) {
        int ww = w + kw - 1;
        if (ww >= 0 && ww < WW_) acc += row[ww] * wgt[(c * 3 + kh) * 3 + kw];
      }
    }
  }

  // L2 norm over the 128 w-values of this row
  float ss = acc * acc;
  #pragma unroll
  for (int m = 16; m >= 1; m >>= 1) ss += __shfl_xor(ss, m, 32);
  __shared__ float red[4];
  int lane = threadIdx.x & 31, wv = threadIdx.x >> 5;
  if (lane == 0) red[wv] = ss;
  __syncthreads();
  float tot = red[0] + red[1] + red[2] + red[3];
  float scale = 1.0f / fmaxf(sqrtf(tot), 1e-12f);

  feat[(((size_t)b * PP) + (size_t)h * WW_ + w) * DD + d] = acc * scale;
}

// ---------------------------------------------------------------------------
// Kernel 2: the appended zero input row -> output row PP is pure bias
// ---------------------------------------------------------------------------
__global__ void bias_row_kernel(const float* __restrict__ ks_b,
                                const float* __restrict__ qs_b,
                                float* __restrict__ flat_ks,
                                float* __restrict__ flat_qs) {
  int t = blockIdx.x * blockDim.x + threadIdx.x;
  if (t >= BB * DD) return;
  int b = t / DD, d = t % DD;
  flat_ks[((size_t)b * P1 + PP) * DD + d] = ks_b[d];
  flat_qs[((size_t)b * P1 + PP) * DD + d] = qs_b[d];
}

// ---------------------------------------------------------------------------
// Kernel 3: dual GEMM via V_WMMA_F32_16X16X4_F32.
//   flat_ks[b,p,:] = feat[b,p,:] @ ks_w.T + ks_b   (and same for qs)
// Workgroup = 288 threads = 9 wave32 (one 16-wide N tile each, 9*16=144 >= 132).
// 64 rows (4 M-tiles) staged into LDS ONCE via GLOBAL_LOAD_ASYNC_TO_LDS_B128
// (ASYNCcnt path), shared across all 9 waves and both weight matrices.
// K loop: unroll(disable) keeps ONE accumulator set (in-place WMMA, no phi
// copies); all 6 fragment loads hoisted to the loop top so they pipeline with
// a single wait before the 8 back-to-back WMMAs.
// ---------------------------------------------------------------------------
__global__ __launch_bounds__(288)
void gemm_dual_kernel(const float* __restrict__ feat,
                      const float* __restrict__ ks_w, const float* __restrict__ ks_b,
                      const float* __restrict__ qs_w, const float* __restrict__ qs_b,
                      float* __restrict__ flat_ks, float* __restrict__ flat_qs) {
  __shared__ __align__(16) float sA[64 * DD];   // 33,792 B of 320 KB/WGP LDS

  int blk = blockIdx.x;
  int rb = blk % (PP / 64);                     // 256 row-blocks per batch
  int b  = blk / (PP / 64);
  int p0 = rb * 64;

  // Async global->LDS staging of the 64x132 fp32 A tile (2112 x b128 chunks).
  // Row stride 528 B = 33*16 -> every 16-B chunk is aligned.
  {
    const char* src = (const char*)(feat + ((size_t)b * PP + p0) * DD);
    unsigned lds_base = (unsigned)(uintptr_t)(&sA[0]);   // low 32 bits = LDS addr
    for (int i = threadIdx.x; i < (64 * DD * 4) / 16; i += 288) {
      unsigned long long gaddr = (unsigned long long)(src + (size_t)i * 16);
      unsigned laddr = lds_base + (unsigned)i * 16;
      asm volatile("global_load_async_to_lds_b128 %0, %1, off"
                   :: "v"(laddr), "v"(gaddr) : "memory");
    }
    asm volatile("s_wait_asynccnt 0x0" ::: "memory");
  }
  __syncthreads();

  int lane = threadIdx.x & 31;
  int wv   = threadIdx.x >> 5;               // N tile id 0..8
  int nb   = wv * 16;
  int half = lane >> 4;                      // K sub-pair select
  int l    = lane & 15;

  int wrow = nb + l;                         // weight row == output column
  bool cvalid = wrow < DD;                   // N padding guard (tile 8)
  int wclamp = cvalid ? wrow : 0;            // clamp -> unconditional loads
  const float* kwp = ks_w + (size_t)wclamp * DD + 2 * half;
  const float* qwp = qs_w + (size_t)wclamp * DD + 2 * half;

  v8f accK[4] = {};
  v8f accQ[4] = {};
  const float* aBase = &sA[l * DD + 2 * half];

  #pragma clang loop unroll(disable)
  for (int k = 0; k < DD; k += 4) {          // 33 K-steps, branch-free body
    // Hoisted fragment loads: 4 outstanding ds_load_b64 + 2 global_load_b64,
    // one wait, then 8 independent in-place WMMAs.
    v2f a0 = *(const v2f*)(aBase + 0 * 16 * DD + k);
    v2f a1 = *(const v2f*)(aBase + 1 * 16 * DD + k);
    v2f a2 = *(const v2f*)(aBase + 2 * 16 * DD + k);
    v2f a3 = *(const v2f*)(aBase + 3 * 16 * DD + k);
    v2f bk = *(const v2f*)(kwp + k);
    v2f bq = *(const v2f*)(qwp + k);
    accK[0] = __builtin_amdgcn_wmma_f32_16x16x4_f32(false, a0, false, bk, (short)0, accK[0], false, false);
    accQ[0] = __builtin_amdgcn_wmma_f32_16x16x4_f32(false, a0, false, bq, (short)0, accQ[0], false, false);
    accK[1] = __builtin_amdgcn_wmma_f32_16x16x4_f32(false, a1, false, bk, (short)0, accK[1], false, false);
    accQ[1] = __builtin_amdgcn_wmma_f32_16x16x4_f32(false, a1, false, bq, (short)0, accQ[1], false, false);
    accK[2] = __builtin_amdgcn_wmma_f32_16x16x4_f32(false, a2, false, bk, (short)0, accK[2], false, false);
    accQ[2] = __builtin_amdgcn_wmma_f32_16x16x4_f32(false, a2, false, bq, (short)0, accQ[2], false, false);
    accK[3] = __builtin_amdgcn_wmma_f32_16x16x4_f32(false, a3, false, bk, (short)0, accK[3], false, false);
    accQ[3] = __builtin_amdgcn_wmma_f32_16x16x4_f32(false, a3, false, bq, (short)0, accQ[3], false, false);
  }

  // C/D layout: VGPR r -> lanes 0-15 (M=r, N=l), lanes 16-31 (M=r+8, N=l)
  int col = nb + l;
  float bkb = cvalid ? ks_b[col] : 0.f;
  float bqb = cvalid ? qs_b[col] : 0.f;
  #pragma unroll
  for (int rt = 0; rt < 4; ++rt) {
    #pragma unroll
    for (int r = 0; r < 8; ++r) {
      int row = p0 + rt * 16 + r + 8 * half;
      if (cvalid) {
        flat_ks[((size_t)b * P1 + row) * DD + col] = accK[rt][r] + bkb;
        flat_qs[((size_t)b * P1 + row) * DD + col] = accQ[rt][r] + bqb;
      }
    }
  }
}

// ---------------------------------------------------------------------------
// Kernel 4: gather rows by indices[1]/indices[2] and dot over D, * D^-0.5.
// One wave32 per (b,n,k); one b128 load per lane per row (rows 16-B aligned),
// lane 0 picks up the 4-float tail. flat_ks+flat_qs = 138 MB -> L2 resident.
// ---------------------------------------------------------------------------
__global__ __launch_bounds__(256)
void gather_dot_kernel(const int* __restrict__ indices,
                       const float* __restrict__ flat_ks,
                       const float* __restrict__ flat_qs,
                       float* __restrict__ out) {
  size_t gw = (size_t)blockIdx.x * (blockDim.x >> 5) + (threadIdx.x >> 5);
  const size_t TOT = (size_t)BB * NN * KK;
  if (gw >= TOT) return;
  int lane = threadIdx.x & 31;
  int b  = (int)(gw / (NN * KK));
  int nk = (int)(gw % (NN * KK));

  int xi = indices[((size_t)1 * BB + b) * (NN * KK) + nk];
  int yi = indices[((size_t)2 * BB + b) * (NN * KK) + nk];

  const float4* xr = (const float4*)(flat_ks + ((size_t)b * P1 + xi) * DD);
  const float4* yr = (const float4*)(flat_qs + ((size_t)b * P1 + yi) * DD);

  float4 xv = xr[lane];
  float4 yv = yr[lane];
  float acc = xv.x * yv.x + xv.y * yv.y + xv.z * yv.z + xv.w * yv.w;
  if (lane == 0) {                      // tail: floats 128..131
    float4 xt = xr[32];
    float4 yt = yr[32];
    acc += xt.x * yt.x + xt.y * yt.y + xt.z * yt.z + xt.w * yt.w;
  }

  #pragma unroll
  for (int m = 16; m >= 1; m >>= 1) acc += __shfl_xor(acc, m, 32);
  if (lane == 0) out[gw] = acc * 0.08703882797784892f;   // 132^-0.5
}

// ---------------------------------------------------------------------------
// kernel_launch: inputs in setup_inputs() order:
//   [0] indices (int32, 3*8*2048*32)  [1] img  [2] conv_w [3] conv_b
//   [4] ks_w    [5] ks_b              [6] qs_w [7] qs_b
// Workspace layout (fp32):
//   feat    : B*P *D = 17,301,504 floats
//   flat_ks : B*P1*D = 17,302,560 floats
//   flat_qs : B*P1*D = 17,302,560 floats   (total ~207.6 MB)
// ---------------------------------------------------------------------------
extern "C" void kernel_launch(void* const* d_in, const int* in_sizes, int n_in,
                              void* d_out, int out_size, void* d_ws, size_t ws_size,
                              hipStream_t stream) {
  const int*   indices = (const int*)  d_in[0];
  const float* img     = (const float*)d_in[1];
  const float* conv_w  = (const float*)d_in[2];
  const float* conv_b  = (const float*)d_in[3];
  const float* ks_w    = (const float*)d_in[4];
  const float* ks_b    = (const float*)d_in[5];
  const float* qs_w    = (const float*)d_in[6];
  const float* qs_b    = (const float*)d_in[7];
  float* out = (float*)d_out;

  float* feat    = (float*)d_ws;
  float* flat_ks = feat    + (size_t)BB * PP * DD;
  float* flat_qs = flat_ks + (size_t)BB * P1 * DD;

  // 1) conv + normalize + transpose
  conv_norm_kernel<<<BB * DD * HH, 128, 0, stream>>>(img, conv_w, conv_b, feat);

  // 2) bias-only row at index P (the appended zero row)
  bias_row_kernel<<<(BB * DD + 255) / 256, 256, 0, stream>>>(ks_b, qs_b, flat_ks, flat_qs);

  // 3) dual fp32 WMMA GEMM with async-to-LDS A staging
  gemm_dual_kernel<<<BB * (PP / 64), 288, 0, stream>>>(
      feat, ks_w, ks_b, qs_w, qs_b, flat_ks, flat_qs);

  // 4) gather + scaled dot
  gather_dot_kernel<<<(int)(((size_t)BB * NN * KK) / 8), 256, 0, stream>>>(
      indices, flat_ks, flat_qs, out);
}